// WeightedGCN_506806141387
// MI455X (gfx1250) — compile-verified
//
#include <hip/hip_runtime.h>

#define D 128

typedef __attribute__((ext_vector_type(2))) float v2f;
typedef __attribute__((ext_vector_type(8))) float v8f;

// ---------------------------------------------------------------------------
// Kernel 1: zero the output accumulator (d_out poisoned to 0xAA by harness)
// ---------------------------------------------------------------------------
__global__ void wgcn_zero_kernel(float4* __restrict__ out, int n4) {
    int i = blockIdx.x * blockDim.x + threadIdx.x;
    if (i < n4) {
        float4 z; z.x = 0.f; z.y = 0.f; z.z = 0.f; z.w = 0.f;
        out[i] = z;
    }
}

// ---------------------------------------------------------------------------
// Kernel 2: h = x @ W^T + b  via V_WMMA_F32_16X16X4_F32
//   block = 256 threads = 8 waves; block owns a 16-row slab of x.
//   wave w owns output cols [16w, 16w+16); K-loop of 32 wmma ops.
//   x tile staged in LDS (8 KB); W (64 KB) read via global (L2 resident).
// ---------------------------------------------------------------------------
__global__ __launch_bounds__(256) void wgcn_gemm_bias_kernel(
    const float* __restrict__ x, const float* __restrict__ W,
    const float* __restrict__ bias, float* __restrict__ h)
{
    __shared__ float Xlds[16 * D];   // 8 KB: 16-row tile of x

    const int tid      = threadIdx.x;
    const int row_base = blockIdx.x * 16;

    // cooperative stage of x tile: 512 float4 across 256 threads
    {
        const float4* Xg = (const float4*)(x + (size_t)row_base * D);
        float4*       Xs = (float4*)Xlds;
        for (int i = tid; i < (16 * D) / 4; i += 256) Xs[i] = Xg[i];
    }
    __syncthreads();

    const int lane     = tid & 31;
    const int wave     = tid >> 5;       // 0..7 -> 16-col tile
    const int col_base = wave * 16;
    const int m        = lane & 15;      // A: row-in-tile | B/C/D: col-in-tile
    const int half     = lane >> 4;      // 0 or 1

    v8f c = {};
    for (int k = 0; k < D; k += 4) {
        // A 16x4 layout: lane holds M=m, K = k + 2*half (+0,+1) -> contiguous
        v2f a = *(const v2f*)(&Xlds[m * D + k + 2 * half]);
        // B 4x16 layout: lane holds N=m, K = k + 2*half (+0,+1)
        // B[k][n] = W[n][k] (h = x @ W^T), W row-major -> contiguous pair
        v2f b = *(const v2f*)(&W[(size_t)(col_base + m) * D + k + 2 * half]);
        c = __builtin_amdgcn_wmma_f32_16x16x4_f32(
                /*neg_a=*/false, a, /*neg_b=*/false, b,
                /*c_mod=*/(short)0, c, /*reuse_a=*/false, /*reuse_b=*/false);
    }

    const float bv = bias[col_base + m];
    // C/D layout: VGPR r -> row (r + 8*half), col m
    #pragma unroll
    for (int r = 0; r < 8; ++r) {
        int row = row_base + r + 8 * half;
        h[(size_t)row * D + col_base + m] = c[r] + bv;
    }
}

// ---------------------------------------------------------------------------
// Kernel 3: out[dst[e]] += h[src[e]] * w[e]   (SpMM scatter-add)
//   one lane per (edge, float4 chunk): 32 lanes cover the 128-f32 row.
//   gather is a coalesced 512B burst per edge; scatter = f32 L2 atomics.
// ---------------------------------------------------------------------------
__global__ __launch_bounds__(256) void wgcn_edge_scatter_kernel(
    const float* __restrict__ h, const float* __restrict__ w,
    const int* __restrict__ src, const int* __restrict__ dst,
    float* __restrict__ out, long long total)
{
    long long gid = (long long)blockIdx.x * blockDim.x + threadIdx.x;
    if (gid >= total) return;
    const int e    = (int)(gid >> 5);
    const int lane = (int)(gid & 31);

    const int   s  = src[e];
    const int   d  = dst[e];
    const float we = w[e];

    const float4 hv = ((const float4*)h)[(size_t)s * (D / 4) + lane];
    float* o = out + (size_t)d * D + lane * 4;
    atomicAdd(o + 0, hv.x * we);
    atomicAdd(o + 1, hv.y * we);
    atomicAdd(o + 2, hv.z * we);
    atomicAdd(o + 3, hv.w * we);
}

// ---------------------------------------------------------------------------
// launcher: inputs in setup_inputs() order: x, w, W, b, src, dst
// ---------------------------------------------------------------------------
extern "C" void kernel_launch(void* const* d_in, const int* in_sizes, int n_in,
                              void* d_out, int out_size, void* d_ws, size_t ws_size,
                              hipStream_t stream) {
    const float* x    = (const float*)d_in[0];
    const float* w    = (const float*)d_in[1];
    const float* W    = (const float*)d_in[2];
    const float* bias = (const float*)d_in[3];
    const int*   src  = (const int*)d_in[4];
    const int*   dst  = (const int*)d_in[5];
    float*       out  = (float*)d_out;
    float*       h    = (float*)d_ws;          // [n_nodes * D] fp32 scratch

    const int n_nodes = in_sizes[0] / D;       // 100000
    const int n_edges = in_sizes[1];           // 3200000

    // 1) zero out
    {
        int n4     = out_size / 4;
        int blocks = (n4 + 255) / 256;
        wgcn_zero_kernel<<<blocks, 256, 0, stream>>>((float4*)out, n4);
    }
    // 2) h = x @ W^T + b   (one 16-row slab per block; n_nodes % 16 == 0)
    {
        int blocks = (n_nodes + 15) / 16;
        wgcn_gemm_bias_kernel<<<blocks, 256, 0, stream>>>(x, W, bias, h);
    }
    // 3) edge gather-scale-scatter
    {
        long long total  = (long long)n_edges * 32;
        long long blocks = (total + 255) / 256;
        wgcn_edge_scatter_kernel<<<(unsigned)blocks, 256, 0, stream>>>(
            h, w, src, dst, out, total);
    }
}